// TorsionLayer_26121991094953
// MI455X (gfx1250) — compile-verified
//
#include <hip/hip_runtime.h>
#include <math.h>

// 180 / pi
#define TORSION_FACTOR 57.295779513082320877f

// Per-point math:
//   phi = acos(z / sqrt(x^2+y^2+z^2));  out0 = -phi * FACTOR
//   psi = acos(x / sqrt(x^2+y^2)); psi = (y<0) ? -psi : psi;  out1 = psi * FACTOR
// Uses v_rsq_f32 (z * rsqrt(s)) instead of a full IEEE divide; preserves the
// reference's NaN behavior (0 * inf = NaN matches 0/0 = NaN).
__device__ __forceinline__ void torsion_point(float x, float y, float z,
                                              float& o0, float& o1) {
    float s2 = fmaf(x, x, y * y);        // x^2 + y^2
    float s3 = fmaf(z, z, s2);           // x^2 + y^2 + z^2
    float rn3 = rsqrtf(s3);              // v_rsq_f32
    float rn2 = rsqrtf(s2);              // v_rsq_f32
    float phi = acosf(z * rn3);          // inlined ocml acos: sqrt + poly
    float psi = acosf(x * rn2);
    psi = (y < 0.0f) ? -psi : psi;       // matches jnp.where(y < 0, -psi, psi)
    o0 = -phi * TORSION_FACTOR;
    o1 = psi  * TORSION_FACTOR;
}

// Vectorized main body: each thread handles 4 points.
//   loads : 3x global_load_b128  (12 floats = 4 AoS-3 points, coalesced)
//   stores: 2x global_store_b128 (8 floats = 4 interleaved [o0,o1] pairs)
__global__ __launch_bounds__(256)
void torsion_vec4_kernel(const float4* __restrict__ in,
                         float4* __restrict__ out,
                         int ngroups) {
    int g = blockIdx.x * blockDim.x + threadIdx.x;
    if (g >= ngroups) return;

    const float4 a = in[3 * g + 0];
    const float4 b = in[3 * g + 1];
    const float4 c = in[3 * g + 2];

    float4 o0, o1;
    // point 0: (a.x, a.y, a.z)   point 1: (a.w, b.x, b.y)
    // point 2: (b.z, b.w, c.x)   point 3: (c.y, c.z, c.w)
    torsion_point(a.x, a.y, a.z, o0.x, o0.y);
    torsion_point(a.w, b.x, b.y, o0.z, o0.w);
    torsion_point(b.z, b.w, c.x, o1.x, o1.y);
    torsion_point(c.y, c.z, c.w, o1.z, o1.w);

    out[2 * g + 0] = o0;
    out[2 * g + 1] = o1;
}

// Scalar tail for npoints % 4 != 0 (not taken for the 1024x8192 shape, but
// keeps the kernel correct for any size).
__global__ __launch_bounds__(64)
void torsion_tail_kernel(const float* __restrict__ in,
                         float* __restrict__ out,
                         int p_start, int npoints) {
    int p = p_start + blockIdx.x * blockDim.x + threadIdx.x;
    if (p >= npoints) return;
    float x = in[3 * p + 0];
    float y = in[3 * p + 1];
    float z = in[3 * p + 2];
    float o0, o1;
    torsion_point(x, y, z, o0, o1);
    out[2 * p + 0] = o0;
    out[2 * p + 1] = o1;
}

extern "C" void kernel_launch(void* const* d_in, const int* in_sizes, int n_in,
                              void* d_out, int out_size, void* d_ws, size_t ws_size,
                              hipStream_t stream) {
    (void)n_in; (void)out_size; (void)d_ws; (void)ws_size;

    const float* x = (const float*)d_in[0];
    float* out = (float*)d_out;

    const long long nelem   = (long long)in_sizes[0]; // B * N * 3
    const long long npoints = nelem / 3;              // B * N
    const long long ngroups = npoints / 4;            // 4 points per thread
    const int       rem     = (int)(npoints - ngroups * 4);

    if (ngroups > 0) {
        const int block = 256;
        const int grid  = (int)((ngroups + block - 1) / block);
        torsion_vec4_kernel<<<grid, block, 0, stream>>>(
            (const float4*)x, (float4*)out, (int)ngroups);
    }
    if (rem > 0) {
        torsion_tail_kernel<<<1, 64, 0, stream>>>(
            x, out, (int)(ngroups * 4), (int)npoints);
    }
}